// HGCN_69672959476265
// MI455X (gfx1250) — compile-verified
//
#include <hip/hip_runtime.h>

#define EMB_D 128
#define TPB   256
#define WPB   8      // waves per block (wave32)

// ---------------------------------------------------------------------------
// CDNA5 native f32 global atomic add (guarantee global_atomic_add_f32, no CAS)
// ---------------------------------------------------------------------------
__device__ __forceinline__ void gadd1(float* p, float x) {
  asm volatile("global_atomic_add_f32 %0, %1, off" :: "v"(p), "v"(x) : "memory");
}

__device__ __forceinline__ void gadd4(float* p, float x, float y, float z, float w) {
  asm volatile(
      "global_atomic_add_f32 %0, %1, off\n\t"
      "global_atomic_add_f32 %0, %2, off offset:4\n\t"
      "global_atomic_add_f32 %0, %3, off offset:8\n\t"
      "global_atomic_add_f32 %0, %4, off offset:12"
      :: "v"(p), "v"(x), "v"(y), "v"(z), "v"(w) : "memory");
}

// ---------------------------------------------------------------------------
// Zero fill (d_ws / d_out are poisoned with 0xAA by the harness)
// ---------------------------------------------------------------------------
__global__ void fill_kernel(float* __restrict__ p, float v, long n) {
  long i  = (long)blockIdx.x * blockDim.x + threadIdx.x;
  long st = (long)gridDim.x * blockDim.x;
  for (; i < n; i += st) p[i] = v;
}

// ---------------------------------------------------------------------------
// Degree accumulation: deg_u[u[e]] += 1, deg_i[i[e]] += 1
// ---------------------------------------------------------------------------
__global__ void degree_kernel(const int* __restrict__ u, const int* __restrict__ it,
                              float* __restrict__ du, float* __restrict__ di, int E) {
  int i  = blockIdx.x * blockDim.x + threadIdx.x;
  int st = gridDim.x * blockDim.x;
  for (; i < E; i += st) {
    gadd1(du + u[i], 1.0f);
    gadd1(di + it[i], 1.0f);
  }
}

// p[i] = 1 / max(p[i], 1)   (covers both inv_deg and deg**(-1.0))
__global__ void invclip_kernel(float* __restrict__ p, int n) {
  int i  = blockIdx.x * blockDim.x + threadIdx.x;
  int st = gridDim.x * blockDim.x;
  for (; i < n; i += st) p[i] = 1.0f / fmaxf(p[i], 1.0f);
}

// ---------------------------------------------------------------------------
// Scatter-add hop:  out[dst[e], :] += (scale ? scale[src[e]] : 1) * in[src[e], :]
// One wave32 per edge; lane l covers dims [4l..4l+3] (coalesced 512B per row).
// Double-buffered async global->LDS staging of the gathered row overlaps the
// random-row gather latency with the atomic-add issue of the previous edge.
// ---------------------------------------------------------------------------
__global__ void scatter_kernel(const float* __restrict__ in,
                               const float* __restrict__ scale,
                               const int* __restrict__ src,
                               const int* __restrict__ dst,
                               float* __restrict__ out, int E) {
  __shared__ float4 smem[2][WPB][32];   // 2 stages x 8 waves x 512B = 8 KB

  const int lane = threadIdx.x & 31;
  const int wave = threadIdx.x >> 5;
  long gw = (long)blockIdx.x * WPB + wave;   // global wave id
  long nw = (long)gridDim.x * WPB;           // total waves
  if (gw >= E) return;

  auto issue = [&](long e, int bf, int& d_out_idx, float& s_out) {
    int s = src[e];
    d_out_idx = dst[e];
    s_out = scale ? scale[s] : 1.0f;
    const float* g = in + (size_t)s * EMB_D + lane * 4;
    unsigned lo = (unsigned)(size_t)(&smem[bf][wave][lane]);   // low 32 bits = LDS offset
    asm volatile("global_load_async_to_lds_b128 %0, %1, off"
                 :: "v"(lo), "v"(g) : "memory");
  };

  long e = gw;
  int buf = 0;
  int dc; float sc;
  issue(e, 0, dc, sc);

  for (;;) {
    long en = e + nw;
    bool hn = en < E;
    int dn = 0; float sn = 0.0f;
    if (hn) {
      issue(en, buf ^ 1, dn, sn);
      // one outstanding (the next stage) allowed; loads retire in order,
      // so asynccnt<=1 means the current buffer's data has landed in LDS
      asm volatile("s_wait_asynccnt 0x1" ::: "memory");
    } else {
      asm volatile("s_wait_asynccnt 0x0" ::: "memory");
    }

    unsigned lo = (unsigned)(size_t)(&smem[buf][wave][lane]);
    float4 v;
    asm volatile("ds_load_b128 %0, %1\n\ts_wait_dscnt 0x0"
                 : "=v"(v) : "v"(lo) : "memory");

    float* p = out + (size_t)dc * EMB_D + lane * 4;
    gadd4(p, v.x * sc, v.y * sc, v.z * sc, v.w * sc);

    if (!hn) break;
    e = en; buf ^= 1; dc = dn; sc = sn;
  }
}

// out[r, :] *= s[r]   (float4 over N*D/4 elements; D=128 -> 32 float4 per row)
__global__ void rowscale_kernel(float4* __restrict__ p, const float* __restrict__ s, long n4) {
  long i  = (long)blockIdx.x * blockDim.x + threadIdx.x;
  long st = (long)gridDim.x * blockDim.x;
  for (; i < n4; i += st) {
    float m = s[i >> 5];
    float4 v = p[i];
    v.x *= m; v.y *= m; v.z *= m; v.w *= m;
    p[i] = v;
  }
}

// ---------------------------------------------------------------------------
// Host orchestration
// ---------------------------------------------------------------------------
extern "C" void kernel_launch(void* const* d_in, const int* in_sizes, int n_in,
                              void* d_out, int out_size, void* d_ws, size_t ws_size,
                              hipStream_t stream) {
  const float* user_emb = (const float*)d_in[0];
  const float* item_emb = (const float*)d_in[1];
  const int*   u_idx    = (const int*)d_in[2];
  const int*   i_idx    = (const int*)d_in[3];

  const int Nu = in_sizes[0] / EMB_D;   // 200000
  const int Ni = in_sizes[1] / EMB_D;   // 100000
  const int E  = in_sizes[2];           // 1000000

  const long nU = (long)Nu * EMB_D;
  const long nI = (long)Ni * EMB_D;

  // workspace layout: inv_u | inv_i | tmp_i (Ni*D) | tmp_u (Nu*D)
  float* inv_u = (float*)d_ws;
  float* inv_i = inv_u + Nu;
  float* tmp_i = inv_i + Ni;
  float* tmp_u = tmp_i + nI;

  // layer-1 intermediates live directly in d_out (recycled for layer-2 output)
  float* out_u = (float*)d_out;
  float* out_i = out_u + nU;

  const int FB = 4096;   // fill blocks
  const int SB = 4096;   // scatter blocks -> 32768 waves, ~31 edges each

  auto fill = [&](float* p, long n) {
    fill_kernel<<<FB, TPB, 0, stream>>>(p, 0.0f, n);
  };
  auto hop = [&](const float* in, const float* sc, const int* s, const int* d, float* o) {
    scatter_kernel<<<SB, TPB, 0, stream>>>(in, sc, s, d, o, E);
  };

  // ---- degrees -> reciprocal (serves both inv_deg and deg**NORM, NORM=-1) ----
  fill(inv_u, Nu);
  fill(inv_i, Ni);
  degree_kernel<<<1024, TPB, 0, stream>>>(u_idx, i_idx, inv_u, inv_i, E);
  invclip_kernel<<<512, TPB, 0, stream>>>(inv_u, Nu);
  invclip_kernel<<<512, TPB, 0, stream>>>(inv_i, Ni);

  // ---- layer 1 (all source-side scalings folded into the next gather) ----
  fill(tmp_i, nI);
  hop(user_emb, nullptr, u_idx, i_idx, tmp_i);          // raw rst_i   (true = *inv_i)
  fill(out_u, nU);
  hop(tmp_i, inv_i, i_idx, u_idx, out_u);               // raw h_u1    (true = *inv_u)
  fill(tmp_u, nU);
  hop(item_emb, nullptr, i_idx, u_idx, tmp_u);          // raw rst_u   (true = *inv_u)
  fill(out_i, nI);
  hop(tmp_u, inv_u, u_idx, i_idx, out_i);               // raw h_i1    (true = *inv_i)

  // ---- layer 2 ----
  fill(tmp_i, nI);
  hop(out_u, inv_u, u_idx, i_idx, tmp_i);               // raw rst_i2  (true = *inv_i)
  fill(out_u, nU);                                      // out_u's last read was above
  hop(tmp_i, inv_i, i_idx, u_idx, out_u);               // raw h_u2    (true = *inv_u)
  fill(tmp_u, nU);
  hop(out_i, inv_i, i_idx, u_idx, tmp_u);               // raw rst_u2  (true = *inv_u)
  fill(out_i, nI);                                      // out_i's last read was above
  hop(tmp_u, inv_u, u_idx, i_idx, out_i);               // raw h_i2    (true = *inv_i)

  // ---- final degree normalizations ----
  rowscale_kernel<<<2048, TPB, 0, stream>>>((float4*)out_u, inv_u, nU / 4);
  rowscale_kernel<<<1024, TPB, 0, stream>>>((float4*)out_i, inv_i, nI / 4);
}